// RGCN_70978629533711
// MI455X (gfx1250) — compile-verified
//
#include <hip/hip_runtime.h>

typedef __attribute__((ext_vector_type(2))) float v2f;
typedef __attribute__((ext_vector_type(8))) float v8f;

// ---------------------------------------------------------------------------
// zero fill, float4-vectorized (grid-stride)
// ---------------------------------------------------------------------------
__global__ __launch_bounds__(256) void zero_f32(float* __restrict__ p, size_t n) {
  size_t n4 = n >> 2;
  float4* p4 = (float4*)p;
  size_t i  = (size_t)blockIdx.x * 256 + threadIdx.x;
  size_t st = (size_t)gridDim.x * 256;
  float4 z = make_float4(0.f, 0.f, 0.f, 0.f);
  for (; i < n4; i += st) p4[i] = z;
  // tail
  size_t t = (n4 << 2) + (size_t)blockIdx.x * 256 + threadIdx.x;
  if (t < n) p[t] = 0.0f;
}

// ---------------------------------------------------------------------------
// fp32 degree histogram: deg[dst[e]] += 1  (hardware global_atomic_add_f32)
// ---------------------------------------------------------------------------
__global__ __launch_bounds__(256) void degree_f32(const int* __restrict__ dst,
                                                  float* __restrict__ deg, int nE) {
  int e = blockIdx.x * 256 + threadIdx.x;
  if (e < nE) unsafeAtomicAdd(&deg[dst[e]], 1.0f);
}

// ---------------------------------------------------------------------------
// segment-mean scatter: out[dst] += X[src] * (1 / max(deg[dst],1)) , d=128
// one wave (32 lanes) per edge, float4 per lane -> 128 floats
// ---------------------------------------------------------------------------
__global__ __launch_bounds__(256) void agg_mean_scatter(
    const float* __restrict__ X, const int* __restrict__ src,
    const int* __restrict__ dst, const float* __restrict__ deg,
    float* __restrict__ out, int nE)
{
  int wave = threadIdx.x >> 5;
  int lane = threadIdx.x & 31;
  int e = blockIdx.x * 8 + wave;
  if (e >= nE) return;
  int s = src[e];
  int d = dst[e];
  float w = 1.0f / fmaxf(deg[d], 1.0f);
  float4 v = ((const float4*)(X + (size_t)s * 128))[lane];
  float* o = out + (size_t)d * 128 + lane * 4;
  unsafeAtomicAdd(o + 0, v.x * w);
  unsafeAtomicAdd(o + 1, v.y * w);
  unsafeAtomicAdd(o + 2, v.z * w);
  unsafeAtomicAdd(o + 3, v.w * w);
}

// ---------------------------------------------------------------------------
// One K=4 WMMA step with compile-time N and compile-time K offset:
//   aBase = A-row + 2*half        (A loads: offset kb*4, b64)
//   bBase = W + 2*half*N + n0+l15 (B loads: offsets kb*N*4, (kb+1)*N*4, b32)
// ---------------------------------------------------------------------------
template <int N>
__device__ __forceinline__ v8f wmma_step(v8f acc, const float* __restrict__ aBase,
                                         const float* __restrict__ bBase, int kb) {
  v2f a, b;
  a[0] = aBase[kb];
  a[1] = aBase[kb + 1];
  b[0] = bBase[kb * N];
  b[1] = bBase[kb * N + N];
  return __builtin_amdgcn_wmma_f32_16x16x4_f32(false, a, false, b,
                                               (short)0, acc, false, false);
}

// ---------------------------------------------------------------------------
// Fused H = act(A1@W1 + b1*[deg1>0] (+ A2@W2 + b2*[deg2>0]))
//   A*: [M,128] f32 row-major, W*: [128,N] row-major (N compile-time), b*: [N]
//   One wave computes one 16x16 tile via v_wmma_f32_16x16x4_f32; K=128 is
//   split into TWO independent accumulator chains per A-matrix to break the
//   D->C RAW serialization. All loads use two base pointers + immediate
//   offsets so the compiler can batch load clauses ahead of the WMMA stream.
//
// WMMA f32 16x16x4 operand layout (ISA 7.12.2):
//   A (16x4): lanes 0-15 -> M=lane, VGPR{0,1}=K{0,1}; lanes 16-31 -> K{2,3}
//   B (4x16): lanes 0-15 -> N=lane, VGPR{0,1}=K{0,1}; lanes 16-31 -> K{2,3}
//   C/D (16x16): VGPR r -> M = r (lanes 0-15) / r+8 (lanes 16-31), N = lane&15
// ---------------------------------------------------------------------------
template <int N, bool HAS2>
__global__ __launch_bounds__(256) void gemm_bias_mask(
    const float* __restrict__ A1, const float* __restrict__ W1p,
    const float* __restrict__ b1p, const float* __restrict__ deg1,
    const float* __restrict__ A2, const float* __restrict__ W2p,
    const float* __restrict__ b2p, const float* __restrict__ deg2,
    float* __restrict__ H, int M, int doRelu,
    int totalTiles, int nTiles)
{
  int wave = threadIdx.x >> 5;
  int lane = threadIdx.x & 31;
  int tile = blockIdx.x * 8 + wave;
  if (tile >= totalTiles) return;

  int mt = tile / nTiles;
  int nt = tile - mt * nTiles;
  int m0 = mt * 16;
  int n0 = nt * 16;

  int half = lane >> 4;      // 0: K{0,1}, 1: K{2,3}
  int l15  = lane & 15;

  int rowA = m0 + l15;
  if (rowA >= M) rowA = M - 1;              // clamp (EXEC must stay all-1s)

  const float* a1Base = A1 + (size_t)rowA * 128 + 2 * half;
  const float* b1Base = W1p + (size_t)(2 * half) * N + n0 + l15;

  v8f acc0 = {}, acc1 = {};
#pragma unroll
  for (int kb = 0; kb < 128; kb += 8) {     // two independent chains
    acc0 = wmma_step<N>(acc0, a1Base, b1Base, kb);
    acc1 = wmma_step<N>(acc1, a1Base, b1Base, kb + 4);
  }

  v8f acc2 = {}, acc3 = {};
  if constexpr (HAS2) {
    const float* a2Base = A2 + (size_t)rowA * 128 + 2 * half;
    const float* b2Base = W2p + (size_t)(2 * half) * N + n0 + l15;
#pragma unroll
    for (int kb = 0; kb < 128; kb += 8) {
      acc2 = wmma_step<N>(acc2, a2Base, b2Base, kb);
      acc3 = wmma_step<N>(acc3, a2Base, b2Base, kb + 4);
    }
  }

  float bias1 = b1p[n0 + l15];
  float bias2 = HAS2 ? b2p[n0 + l15] : 0.0f;

#pragma unroll
  for (int r = 0; r < 8; ++r) {
    int row = m0 + r + half * 8;
    if (row < M) {
      float v = acc0[r] + acc1[r];
      if constexpr (HAS2) v += acc2[r] + acc3[r];
      v += bias1 * (deg1[row] > 0.0f ? 1.0f : 0.0f);
      if constexpr (HAS2) v += bias2 * (deg2[row] > 0.0f ? 1.0f : 0.0f);
      if (doRelu) v = fmaxf(v, 0.0f);
      H[(size_t)row * N + n0 + l15] = v;
    }
  }
}

// ---------------------------------------------------------------------------
// Host-side orchestration
// ---------------------------------------------------------------------------
extern "C" void kernel_launch(void* const* d_in, const int* in_sizes, int n_in,
                              void* d_out, int out_size, void* d_ws, size_t ws_size,
                              hipStream_t stream) {
  const float* embed_user = (const float*)d_in[0];
  const float* embed_item = (const float*)d_in[1];
  const int*   src_uu = (const int*)d_in[2];
  const int*   dst_uu = (const int*)d_in[3];
  const int*   src_ui = (const int*)d_in[4];
  const int*   dst_ui = (const int*)d_in[5];
  const int*   src_iu = (const int*)d_in[6];
  const int*   dst_iu = (const int*)d_in[7];
  const float* W0_uu = (const float*)d_in[8];
  const float* b0_uu = (const float*)d_in[9];
  const float* W0_ui = (const float*)d_in[10];
  const float* b0_ui = (const float*)d_in[11];
  const float* W0_iu = (const float*)d_in[12];
  const float* b0_iu = (const float*)d_in[13];
  const float* W1_uu = (const float*)d_in[14];
  const float* b1_uu = (const float*)d_in[15];
  const float* W1_iu = (const float*)d_in[18];
  const float* b1_iu = (const float*)d_in[19];

  const int NU = in_sizes[0] / 128;
  const int NI = in_sizes[1] / 128;
  const int NE_uu = in_sizes[2];
  const int NE_ui = in_sizes[4];
  const int NE_iu = in_sizes[6];

  // workspace layout (floats):
  //   aggA[NU*128] aggB[NU*128] aggC[NI*128] deg_uu[NU] deg_iu[NU] deg_ui[NI]
  //   h_user[NU*128] h_item[NI*128]
  float* ws = (float*)d_ws;
  float* aggA   = ws;
  float* aggB   = aggA + (size_t)NU * 128;
  float* aggC   = aggB + (size_t)NU * 128;
  float* deg_uu = aggC + (size_t)NI * 128;
  float* deg_iu = deg_uu + NU;
  float* deg_ui = deg_iu + NU;
  float* h_user = deg_ui + NI;
  float* h_item = h_user + (size_t)NU * 128;

  float* out = (float*)d_out;

  // 1) zero agg accumulators + degree arrays (contiguous region)
  {
    size_t nz = (size_t)(2 * NU + NI) * 128 + (size_t)(2 * NU + NI);
    zero_f32<<<8192, 256, 0, stream>>>(aggA, nz);
  }

  // 2) degrees per etype
  degree_f32<<<(NE_uu + 255) / 256, 256, 0, stream>>>(dst_uu, deg_uu, NE_uu);
  degree_f32<<<(NE_iu + 255) / 256, 256, 0, stream>>>(dst_iu, deg_iu, NE_iu);
  degree_f32<<<(NE_ui + 255) / 256, 256, 0, stream>>>(dst_ui, deg_ui, NE_ui);

  // 3) layer-0 segment means of RAW embeddings (linear op commuted past W)
  agg_mean_scatter<<<(NE_uu + 7) / 8, 256, 0, stream>>>(embed_user, src_uu, dst_uu, deg_uu, aggA, NE_uu);
  agg_mean_scatter<<<(NE_iu + 7) / 8, 256, 0, stream>>>(embed_item, src_iu, dst_iu, deg_iu, aggB, NE_iu);
  agg_mean_scatter<<<(NE_ui + 7) / 8, 256, 0, stream>>>(embed_user, src_ui, dst_ui, deg_ui, aggC, NE_ui);

  // 4) h_user = relu(aggA@W0_uu + b0_uu*m_uu + aggB@W0_iu + b0_iu*m_iu)
  {
    int M = NU;
    int nTiles = 128 / 16;
    int totalTiles = ((M + 15) / 16) * nTiles;
    gemm_bias_mask<128, true><<<(totalTiles + 7) / 8, 256, 0, stream>>>(
        aggA, W0_uu, b0_uu, deg_uu, aggB, W0_iu, b0_iu, deg_iu,
        h_user, M, /*relu=*/1, totalTiles, nTiles);
  }
  // 5) h_item = relu(aggC@W0_ui + b0_ui*m_ui)
  {
    int M = NI;
    int nTiles = 128 / 16;
    int totalTiles = ((M + 15) / 16) * nTiles;
    gemm_bias_mask<128, false><<<(totalTiles + 7) / 8, 256, 0, stream>>>(
        aggC, W0_ui, b0_ui, deg_ui, nullptr, nullptr, nullptr, nullptr,
        h_item, M, /*relu=*/1, totalTiles, nTiles);
  }

  // 6) re-zero aggA/aggB for layer-1 accumulation (contiguous)
  zero_f32<<<8192, 256, 0, stream>>>(aggA, (size_t)(2 * NU) * 128);

  // 7) layer-1 segment means of hidden features
  agg_mean_scatter<<<(NE_uu + 7) / 8, 256, 0, stream>>>(h_user, src_uu, dst_uu, deg_uu, aggA, NE_uu);
  agg_mean_scatter<<<(NE_iu + 7) / 8, 256, 0, stream>>>(h_item, src_iu, dst_iu, deg_iu, aggB, NE_iu);

  // 8) out = aggA@W1_uu + b1_uu*m_uu + aggB@W1_iu + b1_iu*m_iu   [NU,16]
  {
    int M = NU;
    int nTiles = 16 / 16;
    int totalTiles = ((M + 15) / 16) * nTiles;
    gemm_bias_mask<16, true><<<(totalTiles + 7) / 8, 256, 0, stream>>>(
        aggA, W1_uu, b1_uu, deg_uu, aggB, W1_iu, b1_iu, deg_iu,
        out, M, /*relu=*/0, totalTiles, nTiles);
  }
}